// MotionDiffusionLoss_59631325938492
// MI455X (gfx1250) — compile-verified
//
#include <hip/hip_runtime.h>
#include <hip/hip_bf16.h>

// MotionDiffusionLoss for MI455X (gfx1250, wave32).
// Bandwidth-bound streaming reduction: one fused pass over pred/target
// (recon + velocity terms share the d = pred-target values), one pass over
// the 4 foot-joint rows of joint_positions. Wave-level sums are computed with
// V_WMMA_F32_16X16X4_F32 (A = lane partials, B = ones -> D[m][n]=p[m]+p[m+16]),
// then one shfl_xor(16). Deterministic: fixed per-block workspace slots, no atomics.

typedef __attribute__((ext_vector_type(2))) float v2f;
typedef __attribute__((ext_vector_type(8))) float v8f;

#define T_DIM      2048
#define N_RECON    18874368.0f   // 64*24*6*2048
#define N_VEL      18865152.0f   // 64*24*6*2047

__device__ __forceinline__ float wave_sum(float p) {
#if __has_builtin(__builtin_amdgcn_wmma_f32_16x16x4_f32)
  // A (16x4 f32, 2 VGPRs): lanes 0-15 -> A[lane][0]=p, lanes 16-31 -> A[lane-16][2]=p.
  // B = all ones, C = 0  =>  D[m][n] = p[m] + p[m+16]  (exact fp32 adds).
  v2f a; a[0] = p;    a[1] = 0.0f;
  v2f b; b[0] = 1.0f; b[1] = 1.0f;
  v8f c = {};
  v8f d = __builtin_amdgcn_wmma_f32_16x16x4_f32(
      /*neg_a=*/false, a, /*neg_b=*/false, b,
      /*c_mod=*/(short)0, c, /*reuse_a=*/false, /*reuse_b=*/false);
  // D layout: VGPR r holds M=r (lanes 0-15) / M=r+8 (lanes 16-31).
  float s = ((d[0] + d[1]) + (d[2] + d[3])) + ((d[4] + d[5]) + (d[6] + d[7]));
  // lanes 0-15 now hold sum(q[0..7]), lanes 16-31 hold sum(q[8..15]).
  s += __shfl_xor(s, 16, 32);
  return s;   // full 32-lane sum in every lane
#else
  #pragma unroll
  for (int off = 16; off > 0; off >>= 1) p += __shfl_xor(p, off, 32);
  return p;
#endif
}

// blockDim.x must be 256 (8 waves). Returns the block sum to every thread.
__device__ __forceinline__ float block_sum_256(float v, float* red) {
  float s = wave_sum(v);
  __syncthreads();                       // allow LDS reuse across calls
  if ((threadIdx.x & 31) == 0) red[threadIdx.x >> 5] = s;
  __syncthreads();
  float t = 0.0f;
  #pragma unroll
  for (int i = 0; i < 8; ++i) t += red[i];
  return t;
}

__device__ __forceinline__ float contact_w(float h) {
  // sigmoid((0.05 - h) * 20) == 1 / (1 + exp((h - 0.05) * 20))
  return 1.0f / (1.0f + __expf((h - 0.05f) * 20.0f));
}

// ---------------- Kernel 1: recon + vel over pred/target -------------------
__global__ void recon_vel_kernel(const float* __restrict__ pred,
                                 const float* __restrict__ target,
                                 float* __restrict__ ws_recon,
                                 float* __restrict__ ws_vel) {
  __shared__ float red[8];
  const int nChunk = 9216 * (T_DIM / 4);          // 4,718,592 float4 chunks
  const int stride = gridDim.x * blockDim.x;
  float racc = 0.0f, vacc = 0.0f;

  for (int k = blockIdx.x * blockDim.x + threadIdx.x; k < nChunk; k += stride) {
    const int e  = k << 2;                         // element index (row-contiguous)
    const int t0 = (k & ((T_DIM / 4) - 1)) << 2;   // position within the T row
    const float4 p = *reinterpret_cast<const float4*>(pred + e);
    const float4 q = *reinterpret_cast<const float4*>(target + e);
    const float d0 = p.x - q.x, d1 = p.y - q.y, d2 = p.z - q.z, d3 = p.w - q.w;
    racc += d0 * d0 + d1 * d1 + d2 * d2 + d3 * d3;
    const float e01 = d1 - d0, e12 = d2 - d1, e23 = d3 - d2;
    vacc += e01 * e01 + e12 * e12 + e23 * e23;
    if (t0 + 4 < T_DIM) {                          // boundary diff into next chunk
      const float d4  = pred[e + 4] - target[e + 4];   // cache-resident reload
      const float e34 = d4 - d3;
      vacc += e34 * e34;
    }
  }

  const float rsum = block_sum_256(racc, red);
  const float vsum = block_sum_256(vacc, red);
  if (threadIdx.x == 0) {
    ws_recon[blockIdx.x] = rsum;
    ws_vel[blockIdx.x]   = vsum;
  }
}

// ---------------- Kernel 2: foot-contact loss over joint_positions ---------
__global__ void foot_kernel(const float* __restrict__ jp,
                            float* __restrict__ ws_num,
                            float* __restrict__ ws_den) {
  __shared__ float red[8];
  const int nChunk = 256 * (T_DIM / 4);            // (64 batches * 4 feet) * 512
  const int stride = gridDim.x * blockDim.x;
  float nacc = 0.0f, dacc = 0.0f;

  for (int k = blockIdx.x * blockDim.x + threadIdx.x; k < nChunk; k += stride) {
    const int pf = k >> 9;                         // (batch, foot) pair 0..255
    const int t0 = (k & 511) << 2;
    const int b  = pf >> 2;
    const int f  = pf & 3;
    const int j  = (f < 2) ? (7 + f) : (8 + f);    // FOOT_IDX = {7, 8, 10, 11}
    const int base = ((b * 24 + j) * 3) * T_DIM;   // c=0 row; +T_DIM per channel

    const float4 x = *reinterpret_cast<const float4*>(jp + base + t0);
    const float4 y = *reinterpret_cast<const float4*>(jp + base + T_DIM + t0);
    const float4 z = *reinterpret_cast<const float4*>(jp + base + 2 * T_DIM + t0);

    const float w0 = contact_w(y.x), w1 = contact_w(y.y);
    const float w2 = contact_w(y.z), w3 = contact_w(y.w);

    { const float vx = x.y - x.x, vz = z.y - z.x, cw = 0.5f * (w0 + w1);
      nacc += (vx * vx + vz * vz) * cw; dacc += cw; }
    { const float vx = x.z - x.y, vz = z.z - z.y, cw = 0.5f * (w1 + w2);
      nacc += (vx * vx + vz * vz) * cw; dacc += cw; }
    { const float vx = x.w - x.z, vz = z.w - z.z, cw = 0.5f * (w2 + w3);
      nacc += (vx * vx + vz * vz) * cw; dacc += cw; }
    if (t0 + 4 < T_DIM) {
      const float x4 = jp[base + t0 + 4];
      const float y4 = jp[base + T_DIM + t0 + 4];
      const float z4 = jp[base + 2 * T_DIM + t0 + 4];
      const float w4 = contact_w(y4);
      const float vx = x4 - x.w, vz = z4 - z.w, cw = 0.5f * (w3 + w4);
      nacc += (vx * vx + vz * vz) * cw; dacc += cw;
    }
  }

  const float nsum = block_sum_256(nacc, red);
  const float dsum = block_sum_256(dacc, red);
  if (threadIdx.x == 0) {
    ws_num[blockIdx.x] = nsum;
    ws_den[blockIdx.x] = dsum;
  }
}

// ---------------- Kernel 3: deterministic final combine --------------------
__global__ void finalize_kernel(const float* __restrict__ ws_recon,
                                const float* __restrict__ ws_vel,
                                const float* __restrict__ ws_num,
                                const float* __restrict__ ws_den,
                                float* __restrict__ out) {
  __shared__ float red[8];
  float r = 0.0f, v = 0.0f;
  for (int i = threadIdx.x; i < 2048; i += 256) { r += ws_recon[i]; v += ws_vel[i]; }
  float n = ws_num[threadIdx.x];
  float d = ws_den[threadIdx.x];

  r = block_sum_256(r, red);
  v = block_sum_256(v, red);
  n = block_sum_256(n, red);
  d = block_sum_256(d, red);

  if (threadIdx.x == 0) {
    const float recon = r / N_RECON;
    const float vel   = v / N_VEL;
    const float foot  = n / (d + 1e-8f);
    out[0] = recon + 0.2f * vel + 0.1f * foot;
  }
}

extern "C" void kernel_launch(void* const* d_in, const int* in_sizes, int n_in,
                              void* d_out, int out_size, void* d_ws, size_t ws_size,
                              hipStream_t stream) {
  const float* pred   = (const float*)d_in[0];
  const float* target = (const float*)d_in[1];
  const float* jp     = (const float*)d_in[2];
  float* out = (float*)d_out;
  float* ws  = (float*)d_ws;

  float* ws_recon = ws;             // 2048 floats
  float* ws_vel   = ws + 2048;      // 2048 floats
  float* ws_num   = ws + 4096;      // 256 floats
  float* ws_den   = ws + 4096 + 256;// 256 floats

  recon_vel_kernel<<<2048, 256, 0, stream>>>(pred, target, ws_recon, ws_vel);
  foot_kernel<<<256, 256, 0, stream>>>(jp, ws_num, ws_den);
  finalize_kernel<<<1, 256, 0, stream>>>(ws_recon, ws_vel, ws_num, ws_den, out);
}